// ParallelPLI_71889162600690
// MI455X (gfx1250) — compile-verified
//
#include <hip/hip_runtime.h>
#include <stdint.h>

typedef __attribute__((ext_vector_type(2))) float v2f;
typedef __attribute__((ext_vector_type(8))) float v8f;
typedef __attribute__((ext_vector_type(4))) unsigned int u32x4;
typedef __attribute__((ext_vector_type(4))) int i32x4;
typedef __attribute__((ext_vector_type(8))) int i32x8;

constexpr int T_STEPS = 1024;
constexpr int COLS    = 16384;            // B * D
constexpr int CHUNK   = 16;
constexpr int NCHUNK  = T_STEPS / CHUNK;  // 64
constexpr int NBUF    = 4;                // TDM pipeline depth (tiles in flight)
constexpr int WPB     = 8;                // waves per block

#if defined(__has_builtin)
#  if __has_builtin(__builtin_amdgcn_tensor_load_to_lds)
#    define HAVE_TDM 1
#  endif
#endif
#ifndef HAVE_TDM
#  define HAVE_TDM 0
#endif

// Issue one TDM descriptor: 16x16 fp32 tile, row stride COLS, global -> LDS.
__device__ __forceinline__ void tdm_load_tile(uint32_t lds_addr, uint64_t gaddr)
{
#if HAVE_TDM
    u32x4 g0;
    g0.x = 1u;                                    // count=1 (valid user D#)
    g0.y = lds_addr;                              // lds_addr[31:0] (bytes)
    g0.z = (uint32_t)gaddr;                       // global_addr[31:0]
    g0.w = (uint32_t)(gaddr >> 32) | (2u << 30);  // global_addr[56:32] | type=2

    i32x8 g1;
    g1[0] = (int)(2u << 16);                                  // data_size=4B, no multicast
    g1[1] = (int)(((uint32_t)COLS & 0xFFFFu) << 16);          // tensor_dim0 lo16
    g1[2] = (int)(((uint32_t)COLS >> 16) |
                  (((uint32_t)T_STEPS & 0xFFFFu) << 16));     // dim0 hi | tensor_dim1 lo
    g1[3] = (int)(((uint32_t)T_STEPS >> 16) | (16u << 16));   // dim1 hi | tile_dim0=16
    g1[4] = 16;                                               // tile_dim1=16, tile_dim2=0
    g1[5] = COLS;                                             // tensor_dim0_stride lo32
    g1[6] = 0;                                                // stride hi | dim1_stride lo
    g1[7] = 0;

    i32x4 gz = {0, 0, 0, 0};
#  if __has_include(<hip/amd_detail/amd_gfx1250_TDM.h>)
    i32x8 gz8 = {0, 0, 0, 0, 0, 0, 0, 0};
    __builtin_amdgcn_tensor_load_to_lds(g0, g1, gz, gz, gz8, 0);
#  else
    __builtin_amdgcn_tensor_load_to_lds(g0, g1, gz, gz, 0);
#  endif
#else
    (void)lds_addr; (void)gaddr;
#endif
}

// One wave32 owns a 16-column tile across all T=1024 steps, in 64 chunks of
// 16. TDM streams each 16x16 input tile into a per-wave LDS ring (depth 4);
// chunk combine is y_local = L16 @ X via 4x V_WMMA_F32_16X16X4_F32 with the
// cross-chunk carry folded into the C seed.
__global__ __launch_bounds__(WPB * 32)
void decay_scan_wmma_tdm(const float* __restrict__ x,
                         const float* __restrict__ beta_raw,
                         float* __restrict__ y)
{
    __shared__ float smem[WPB * NBUF * 256];   // 32 KB: 1KB tile per buffer

    const int lane  = threadIdx.x & 31;
    const int wslot = threadIdx.x >> 5;                          // wave in block
    const int wid   = (blockIdx.x * blockDim.x + threadIdx.x) >> 5;
    const int nloc  = lane & 15;          // column within tile (N index)
    const int hi    = lane >> 4;          // lane half selects K / M striping
    const int col   = wid * 16 + nloc;

    // beta = sigmoid(raw scalar), wave-uniform; powers beta^0..beta^16
    const float beta = 1.0f / (1.0f + __expf(-beta_raw[0]));
    float pw[17];
    pw[0] = 1.0f;
#pragma unroll
    for (int i = 1; i <= 16; ++i) pw[i] = pw[i - 1] * beta;

    // A = L16 lower-triangular decay matrix in 4 K-slices (16x4 f32 each).
    v2f A[4];
#pragma unroll
    for (int s = 0; s < 4; ++s) {
        const int k0 = s * 4 + 2 * hi;
        const int e0 = nloc - k0;
        const int e1 = nloc - (k0 + 1);
        A[s].x = (e0 >= 0) ? pw[e0] : 0.0f;
        A[s].y = (e1 >= 0) ? pw[e1] : 0.0f;
    }

    // Per-wave LDS ring base (generic-pointer low 32 bits == LDS byte offset).
    float* ring = &smem[wslot * NBUF * 256];
    const uint32_t lds_base = (uint32_t)(uintptr_t)(void*)ring;
    const uint64_t gbase = (uint64_t)(uintptr_t)x + (uint64_t)(wid * 16) * 4u;
    const uint64_t gstep = (uint64_t)CHUNK * (uint64_t)COLS * 4u;  // 1 MB / chunk

#if HAVE_TDM
#pragma unroll
    for (int c = 0; c < NBUF; ++c)
        tdm_load_tile(lds_base + (uint32_t)(c & (NBUF - 1)) * 1024u,
                      gbase + (uint64_t)c * gstep);
#endif

    float* yc = y + col;
    float carry = 0.0f;   // y[t0-1] for this column, duplicated in both halves

    auto process = [&](int c) {
        const float* bp = ring + (c & (NBUF - 1)) * 256;
        v2f B[4];
#pragma unroll
        for (int s = 0; s < 4; ++s) {
            const int r = s * 4 + 2 * hi;
#if HAVE_TDM
            B[s].x = bp[r * 16 + nloc];            // ds_load from staged tile
            B[s].y = bp[(r + 1) * 16 + nloc];
#else
            B[s].x = x[(size_t)(c * CHUNK + r) * COLS + col];
            B[s].y = x[(size_t)(c * CHUNK + r + 1) * COLS + col];
#endif
        }
#if HAVE_TDM
        // Reads landed in VGPRs before this buffer gets re-targeted by TDM.
        asm volatile("s_wait_dscnt 0x0" ::: "memory");
        if (c + NBUF < NCHUNK)
            tdm_load_tile(lds_base + (uint32_t)((c + NBUF) & (NBUF - 1)) * 1024u,
                          gbase + (uint64_t)(c + NBUF) * gstep);
#endif
        // Seed C with carry: y[t0+i] += beta^(i+1) * carry.  C/D layout:
        // lane = N, VGPR v = row (v + 8*hi).
        v8f d;
#pragma unroll
        for (int v = 0; v < 8; ++v) d[v] = pw[v + 8 * hi + 1] * carry;

#pragma unroll
        for (int s = 0; s < 4; ++s) {
            d = __builtin_amdgcn_wmma_f32_16x16x4_f32(
                    false, A[s], false, B[s], (short)0, d, false, false);
        }

        const int t0 = c * CHUNK;
#pragma unroll
        for (int v = 0; v < 8; ++v)
            yc[(size_t)(t0 + v + 8 * hi) * COLS] = d[v];

        carry = __shfl(d[7], 16 + nloc, 32);   // broadcast row 15
    };

    int c = 0;
    for (; c < NCHUNK - NBUF; ++c) {           // steady state: depth-NBUF pipe
#if HAVE_TDM
        __builtin_amdgcn_s_wait_tensorcnt(NBUF - 1);
#endif
        process(c);
    }
    for (; c < NCHUNK; ++c) {                  // drain: no further issues
#if HAVE_TDM
        __builtin_amdgcn_s_wait_tensorcnt(0);
#endif
        process(c);
    }
}

extern "C" void kernel_launch(void* const* d_in, const int* in_sizes, int n_in,
                              void* d_out, int out_size, void* d_ws, size_t ws_size,
                              hipStream_t stream) {
    (void)in_sizes; (void)n_in; (void)d_ws; (void)ws_size; (void)out_size;
    const float* x        = (const float*)d_in[0];
    const float* beta_raw = (const float*)d_in[1];
    float*       y        = (float*)d_out;

    // 16384 columns / 16 per wave = 1024 waves; 8 waves (256 thr) per block.
    const int blocks = (COLS / 16) / WPB;    // 128
    decay_scan_wmma_tdm<<<blocks, WPB * 32, 0, stream>>>(x, beta_raw, y);
}